// StructModel_77962246357359
// MI455X (gfx1250) — compile-verified
//
#include <hip/hip_runtime.h>

// ---------------- constants (match reference) ----------------
constexpr int N_NODES = 8192;
constexpr int V       = 8;
constexpr int E_EDGES = N_NODES - 1;          // 8191
constexpr int BATCH   = 64;
constexpr int D_IN    = 64;
constexpr int NV      = N_NODES * V;          // 65536
constexpr int NUM_POT = NV + E_EDGES * V * V; // 589760
constexpr int MP_ITRS = 10;

// chain blocking
constexpr int SEG     = 128;                  // core edges/nodes per block
constexpr int HALO    = 16;                   // >= MP_ITRS, safe halo
constexpr int EXT_MAX = SEG + 2 * HALO;       // 160
constexpr int TPAD    = 68;                   // theta_p LDS stride (floats), float4-friendly, conflict-free
constexpr int NSEG    = N_NODES / SEG;        // 64

typedef float v2f __attribute__((ext_vector_type(2)));
typedef float v8f __attribute__((ext_vector_type(8)));

// ================= Kernel 1: potentials GEMM via f32 WMMA =================
// theta[b, c] = inp[b,:] . W[:,c] + bias[c], c in [0, NUM_POT)
// block = 128 threads (4 waves); each wave: 16 columns x 64 rows (4 WMMA M-tiles)
__global__ __launch_bounds__(128) void pots_gemm_kernel(
    const float* __restrict__ inp,
    const float* __restrict__ Wu, const float* __restrict__ bu,
    const float* __restrict__ Wp, const float* __restrict__ bp,
    float* __restrict__ theta)
{
    __shared__ float Alds[BATCH * D_IN];      // 16 KB: whole inp matrix
    for (int t = threadIdx.x; t < (BATCH * D_IN) / 4; t += 128)
        ((float4*)Alds)[t] = ((const float4*)inp)[t];
    __syncthreads();

    const int wave = threadIdx.x >> 5;
    const int lane = threadIdx.x & 31;
    const int half = lane >> 4;               // 0: K pair {k0,k0+1}, 1: {k0+2,k0+3}
    const int idx  = lane & 15;
    const int col  = blockIdx.x * 64 + wave * 16 + idx;   // tiles of 16/64 never straddle NV

    const float* W; const float* bias; int cl; long Cstride;
    if (col < NV) { W = Wu; bias = bu; cl = col;      Cstride = NV; }
    else          { W = Wp; bias = bp; cl = col - NV; Cstride = (long)E_EDGES * 64; }

    v8f acc0 = {}, acc1 = {}, acc2 = {}, acc3 = {};

    for (int ks = 0; ks < 16; ++ks) {         // K = 64 in steps of 4
        const int kb = ks * 4 + 2 * half;
        v2f bf;
        bf.x = W[(long)(kb + 0) * Cstride + cl];
        bf.y = W[(long)(kb + 1) * Cstride + cl];

        v2f a0 = *(const v2f*)&Alds[( 0 + idx) * D_IN + kb];
        v2f a1 = *(const v2f*)&Alds[(16 + idx) * D_IN + kb];
        v2f a2 = *(const v2f*)&Alds[(32 + idx) * D_IN + kb];
        v2f a3 = *(const v2f*)&Alds[(48 + idx) * D_IN + kb];

        acc0 = __builtin_amdgcn_wmma_f32_16x16x4_f32(false, a0, false, bf, (short)0, acc0, false, false);
        acc1 = __builtin_amdgcn_wmma_f32_16x16x4_f32(false, a1, false, bf, (short)0, acc1, false, false);
        acc2 = __builtin_amdgcn_wmma_f32_16x16x4_f32(false, a2, false, bf, (short)0, acc2, false, false);
        acc3 = __builtin_amdgcn_wmma_f32_16x16x4_f32(false, a3, false, bf, (short)0, acc3, false, false);
    }

    const float bv = bias[cl];
    const int rbase = half * 8;
    #pragma unroll
    for (int r = 0; r < 8; ++r) {
        theta[(long)( 0 + rbase + r) * NUM_POT + col] = acc0[r] + bv;
        theta[(long)(16 + rbase + r) * NUM_POT + col] = acc1[r] + bv;
        theta[(long)(32 + rbase + r) * NUM_POT + col] = acc2[r] + bv;
        theta[(long)(48 + rbase + r) * NUM_POT + col] = acc3[r] + bv;
    }
}

// ================= Kernel 2: halo-blocked chain BP + beliefs =================
// block = (segment, batch item); 256 threads = 32 edges x 8 vals
__global__ __launch_bounds__(256) void mp_kernel(
    const float* __restrict__ theta, float* __restrict__ out)
{
    const int b    = blockIdx.y;
    const int s    = blockIdx.x * SEG;
    const int e_lo = (s - HALO > 0) ? (s - HALO) : 0;
    const int e_hi = (s + SEG + HALO < E_EDGES) ? (s + SEG + HALO) : E_EDGES;
    const int ecnt = e_hi - e_lo;             // <= 160
    const int ncnt = ecnt + 1;                // nodes e_lo .. e_hi

    __shared__ float th_u[(EXT_MAX + 1) * V]; // [ncnt][8]
    __shared__ float th_p[EXT_MAX * TPAD];    // [ecnt][64] padded to 68
    __shared__ float mf[EXT_MAX * V];
    __shared__ float mb[EXT_MAX * V];

    const float* thb = theta + (long)b * NUM_POT;

    for (int t = threadIdx.x; t < ncnt * V; t += 256)
        th_u[t] = thb[e_lo * V + t];

    const float4* pp = (const float4*)(thb + NV + (long)e_lo * 64);
    for (int q = threadIdx.x; q < ecnt * 16; q += 256) {
        const int e = q >> 4, r = q & 15;
        float4 v = pp[q];
        float* dst = &th_p[e * TPAD + r * 4];
        dst[0] = v.x; dst[1] = v.y; dst[2] = v.z; dst[3] = v.w;
    }
    for (int t = threadIdx.x; t < EXT_MAX * V; t += 256) { mf[t] = 0.f; mb[t] = 0.f; }
    __syncthreads();

    const int j    = threadIdx.x & 7;         // value index
    const int eoff = threadIdx.x >> 3;        // 0..31

    for (int it = 0; it < MP_ITRS; ++it) {
        float nfv[5], nbv[5];
        #pragma unroll
        for (int k = 0; k < 5; ++k) {
            const int le = eoff + 32 * k;
            if (le >= ecnt) { nfv[k] = 0.f; nbv[k] = 0.f; continue; }
            float t0[8];
            // m_fwd_new[e][j] = lse_i( th_p[e][i][j] + th_u[e][i] + m_fwd[e-1][i] )
            float mx = -1e30f;
            #pragma unroll
            for (int i = 0; i < 8; ++i) {
                const float nl = th_u[le * V + i] + (le > 0 ? mf[(le - 1) * V + i] : 0.f);
                const float tv = th_p[le * TPAD + i * V + j] + nl;
                t0[i] = tv; mx = fmaxf(mx, tv);
            }
            float sm = 0.f;
            #pragma unroll
            for (int i = 0; i < 8; ++i) sm += __expf(t0[i] - mx);
            float newf = mx + __logf(sm);
            // m_bwd_new[e][j(row)] = lse_c( th_p[e][j][c] + th_u[e+1][c] + m_bwd[e+1][c] )
            float mx2 = -1e30f;
            #pragma unroll
            for (int c = 0; c < 8; ++c) {
                const float nr = th_u[(le + 1) * V + c] + (le + 1 < ecnt ? mb[(le + 1) * V + c] : 0.f);
                const float tv = th_p[le * TPAD + j * V + c] + nr;
                t0[c] = tv; mx2 = fmaxf(mx2, tv);
            }
            float s2 = 0.f;
            #pragma unroll
            for (int c = 0; c < 8; ++c) s2 += __expf(t0[c] - mx2);
            float newb = mx2 + __logf(s2);
            // gauge normalization: subtract mean over the 8 lanes sharing this edge
            float sf = newf, sb = newb;
            #pragma unroll
            for (int m = 1; m < 8; m <<= 1) { sf += __shfl_xor(sf, m, 8); sb += __shfl_xor(sb, m, 8); }
            nfv[k] = newf - sf * 0.125f;
            nbv[k] = newb - sb * 0.125f;
        }
        __syncthreads();
        #pragma unroll
        for (int k = 0; k < 5; ++k) {
            const int le = eoff + 32 * k;
            if (le < ecnt) { mf[le * V + j] = nfv[k]; mb[le * V + j] = nbv[k]; }
        }
        __syncthreads();
    }

    float* outb = out + (long)b * NUM_POT;

    // ---- unary beliefs: softmax(th_u[n] + m_fwd[n-1] + m_bwd[n]) over V ----
    const int n_hi = (s + SEG < N_NODES) ? (s + SEG) : N_NODES;
    #pragma unroll
    for (int k = 0; k < 4; ++k) {
        const int n = s + eoff + 32 * k;
        if (n >= n_hi) continue;
        const int ln = n - e_lo;
        float inc = th_u[ln * V + j]
                  + (ln > 0    ? mf[(ln - 1) * V + j] : 0.f)
                  + (ln < ecnt ? mb[ln * V + j]       : 0.f);
        float mx = inc;
        for (int m = 1; m < 8; m <<= 1) mx = fmaxf(mx, __shfl_xor(mx, m, 8));
        float ex = __expf(inc - mx);
        float sm = ex;
        for (int m = 1; m < 8; m <<= 1) sm += __shfl_xor(sm, m, 8);
        outb[n * V + j] = ex / sm;
    }

    // ---- pair beliefs: softmax over 64 of th_p + n_l[:,None] + n_r[None,:] ----
    const int ce_hi = (s + SEG < E_EDGES) ? (s + SEG) : E_EDGES;
    #pragma unroll
    for (int k = 0; k < 4; ++k) {
        const int e = s + eoff + 32 * k;
        if (e >= ce_hi) continue;
        const int le = e - e_lo;
        const float nr = th_u[(le + 1) * V + j] + (le + 1 < ecnt ? mb[(le + 1) * V + j] : 0.f);
        float t0[8]; float mx = -1e30f;
        #pragma unroll
        for (int i = 0; i < 8; ++i) {
            const float nl = th_u[le * V + i] + (le > 0 ? mf[(le - 1) * V + i] : 0.f);
            const float tv = th_p[le * TPAD + i * V + j] + nl + nr;
            t0[i] = tv; mx = fmaxf(mx, tv);
        }
        for (int m = 1; m < 8; m <<= 1) mx = fmaxf(mx, __shfl_xor(mx, m, 8));
        float sm = 0.f;
        #pragma unroll
        for (int i = 0; i < 8; ++i) { t0[i] = __expf(t0[i] - mx); sm += t0[i]; }
        for (int m = 1; m < 8; m <<= 1) sm += __shfl_xor(sm, m, 8);
        const float inv = 1.0f / sm;
        float* po = outb + NV + (long)e * 64;
        #pragma unroll
        for (int i = 0; i < 8; ++i) po[i * V + j] = t0[i] * inv;
    }
}

// ================= launch =================
extern "C" void kernel_launch(void* const* d_in, const int* in_sizes, int n_in,
                              void* d_out, int out_size, void* d_ws, size_t ws_size,
                              hipStream_t stream) {
    (void)in_sizes; (void)n_in; (void)out_size; (void)ws_size;
    const float* inp = (const float*)d_in[0];
    const float* Wu  = (const float*)d_in[1];
    const float* bu  = (const float*)d_in[2];
    const float* Wp  = (const float*)d_in[3];
    const float* bp  = (const float*)d_in[4];
    float* out   = (float*)d_out;
    float* theta = (float*)d_ws;   // needs BATCH * NUM_POT * 4 = ~151 MB scratch

    pots_gemm_kernel<<<dim3(NUM_POT / 64), 128, 0, stream>>>(inp, Wu, bu, Wp, bp, theta);
    mp_kernel<<<dim3(NSEG, BATCH), 256, 0, stream>>>(theta, out);
}